// MDTA_33612414059016
// MI455X (gfx1250) — compile-verified
//
#include <hip/hip_runtime.h>
#include <math.h>

// ---------------------------------------------------------------------------
// MDTA (Restormer transposed channel attention) for MI455X / gfx1250.
// Heavy math on V_WMMA_F32_16X16X4_F32 (native fp32 WMMA, wave32);
// GEMM tiles staged with CDNA5 async global->LDS copies (ASYNCcnt),
// double-buffered so stage s+1 is in flight while WMMAs consume stage s.
//
// Dims: B=8, C=384, HEADS=8, HD=48, H=W=64, N=HW=4096, C3=3C=1152.
// ---------------------------------------------------------------------------

typedef float v2f __attribute__((ext_vector_type(2)));
typedef float v8f __attribute__((ext_vector_type(8)));

#define WMMA_F32(a, b, c) \
  __builtin_amdgcn_wmma_f32_16x16x4_f32(false, (a), false, (b), (short)0, (c), false, false)

// CDNA5 async global->LDS 16-byte copy (VGLOBAL encoding, tracked by ASYNCcnt).
// lds_off: byte offset into the workgroup's LDS (low 32 bits of flat address).
__device__ __forceinline__ void async_copy_b128(unsigned lds_off, const void* g) {
  asm volatile("global_load_async_to_lds_b128 %0, %1, off"
               :: "v"(lds_off), "v"((unsigned long long)(uintptr_t)g)
               : "memory");
}
__device__ __forceinline__ unsigned lds_off(const void* p) {
  return (unsigned)(uintptr_t)p;   // flat LDS address low 32 bits == LDS offset
}

// ---------------------------------------------------------------------------
// Generic batched GEMM:  Y[b] = W[M,K] @ X[b][K,N]   (all row-major, fp32)
// Block tile 64(M) x 128(N), 8 waves arranged 4(M) x 2(N); each wave owns a
// 16x64 strip (4 accumulators). K staged through double-buffered LDS in steps
// of 16 via async copies. A kept row-major with +4 pad (stride 20 floats) so
// the strided 16-lane fragment read is bank-conflict free.
// grid = (N/128, M/64, batch), block = 256.
// ---------------------------------------------------------------------------
__global__ __launch_bounds__(256) void gemm_wmma_f32(
    const float* __restrict__ W, const float* __restrict__ X,
    float* __restrict__ Y, int M, int K, int N)
{
  __shared__ __align__(16) float As[2][64][20];    // [buf][m][k] (+pad)
  __shared__ __align__(16) float Bs[2][16][128];   // [buf][k][n]

  const int tid  = threadIdx.x;
  const int lane = tid & 31;
  const int wave = tid >> 5;
  const int wm   = wave >> 1;          // 0..3 (M sub-tile)
  const int wn   = wave & 1;           // 0..1 (N half)
  const int l15  = lane & 15;
  const int hi   = lane >> 4;          // 0 or 1
  const int kh   = hi << 1;            // K half-offset: 0 or 2

  const int n0 = blockIdx.x * 128;
  const int m0 = blockIdx.y * 64;
  const int b  = blockIdx.z;

  const float* Xb = X + (size_t)b * K * N;
  float*       Yb = Y + (size_t)b * M * N;

  // Stage-copy mapping: A tile 64x16 floats = 256 x 16B chunks (1/thread);
  // B tile 16x128 floats = 512 x 16B chunks (2/thread).
  const int arow = tid >> 2;
  const int acol = (tid & 3) * 4;

  auto stage = [&](int buf, int kb) {
    async_copy_b128(lds_off(&As[buf][arow][acol]),
                    W + (size_t)(m0 + arow) * K + kb + acol);
#pragma unroll
    for (int j = 0; j < 2; ++j) {
      const int c   = tid + j * 256;
      const int r   = c >> 5;
      const int col = (c & 31) * 4;
      async_copy_b128(lds_off(&Bs[buf][r][col]),
                      Xb + (size_t)(kb + r) * N + n0 + col);
    }
  };

  v8f acc[4] = {{}, {}, {}, {}};

  stage(0, 0);
  const int steps = K / 16;
  for (int s = 0; s < steps; ++s) {
    const int  buf  = s & 1;
    const bool more = (s + 1) < steps;
    if (more) {
      stage(buf ^ 1, (s + 1) * 16);                 // buf^1 free since iter s-1
      asm volatile("s_wait_asynccnt 0x3" ::: "memory");  // stage s landed
    } else {
      asm volatile("s_wait_asynccnt 0x0" ::: "memory");
    }
    __syncthreads();

#pragma unroll
    for (int kk = 0; kk < 16; kk += 4) {
      const int kf = kk + kh;
      v2f a;
      a.x = As[buf][wm * 16 + l15][kf];
      a.y = As[buf][wm * 16 + l15][kf + 1];
#pragma unroll
      for (int j = 0; j < 4; ++j) {
        v2f bb;
        bb.x = Bs[buf][kf][wn * 64 + j * 16 + l15];
        bb.y = Bs[buf][kf + 1][wn * 64 + j * 16 + l15];
        acc[j] = WMMA_F32(a, bb, acc[j]);
      }
    }
    __syncthreads();
  }

  // C/D layout: VGPR r -> M = r (lanes 0-15) or r+8 (lanes 16-31), N = lane&15
  const int rbase = m0 + wm * 16 + hi * 8;
  const int cbase = n0 + wn * 64 + l15;
#pragma unroll
  for (int r = 0; r < 8; ++r)
#pragma unroll
    for (int j = 0; j < 4; ++j)
      Yb[(size_t)(rbase + r) * N + cbase + j * 16] = acc[j][r];
}

// ---------------------------------------------------------------------------
// Depthwise 3x3 "SAME" correlation (zero pad), groups = channels.
// One thread per output element. in/out: [B][1152][64][64], w: [1152][9].
// ---------------------------------------------------------------------------
__global__ __launch_bounds__(256) void dwconv3x3(
    const float* __restrict__ in, const float* __restrict__ w,
    float* __restrict__ out)
{
  const int idx = blockIdx.x * 256 + threadIdx.x;   // < 8*1152*4096
  const int x  = idx & 63;
  const int y  = (idx >> 6) & 63;
  const int bc = idx >> 12;                          // b*1152 + c
  const int c  = bc % 1152;

  const float* ip = in + (size_t)bc * 4096;
  const float* wp = w + c * 9;

  float s = 0.0f;
#pragma unroll
  for (int dy = 0; dy < 3; ++dy) {
    const int yy = y + dy - 1;
    if (yy < 0 || yy > 63) continue;
#pragma unroll
    for (int dx = 0; dx < 3; ++dx) {
      const int xx = x + dx - 1;
      if (xx < 0 || xx > 63) continue;
      s += ip[yy * 64 + xx] * wp[dy * 3 + dx];
    }
  }
  out[idx] = s;
}

// ---------------------------------------------------------------------------
// Per-row L2 norm over 4096 spatial elements for q (ch 0..383) and k
// (ch 384..767) of the depthwise output. inv[b*768 + ch] = 1/max(||row||,eps)
// grid = 8*768 blocks of 256.
// ---------------------------------------------------------------------------
__global__ __launch_bounds__(256) void l2norm_rows(
    const float* __restrict__ dw, float* __restrict__ inv)
{
  const int row = blockIdx.x;           // 0..8*768-1
  const int b   = row / 768;
  const int ch  = row % 768;            // 0..383 q, 384..767 k
  const float* p = dw + ((size_t)b * 1152 + ch) * 4096;

  float ss = 0.0f;
  for (int i = threadIdx.x; i < 4096; i += 256) {
    const float v = p[i];
    ss += v * v;
  }
  __shared__ float red[256];
  red[threadIdx.x] = ss;
  __syncthreads();
  for (int s = 128; s > 0; s >>= 1) {
    if (threadIdx.x < s) red[threadIdx.x] += red[threadIdx.x + s];
    __syncthreads();
  }
  if (threadIdx.x == 0) inv[row] = 1.0f / fmaxf(sqrtf(red[0]), 1e-12f);
}

// ---------------------------------------------------------------------------
// attn = softmax( (q_n @ k_n^T) * temperature )  per (b,head), [48][48].
// GEMM M=N=48, K=4096 via WMMA f32 16x16x4: 3 waves x 3 N-tiles x 1024 k-iters.
// grid = 64 (b*heads), block = 96.
// ---------------------------------------------------------------------------
__global__ __launch_bounds__(96) void attn_softmax(
    const float* __restrict__ dw, const float* __restrict__ inv,
    const float* __restrict__ temp, float* __restrict__ attn)
{
  const int bh   = blockIdx.x;
  const int b    = bh >> 3;
  const int head = bh & 7;
  const int tid  = threadIdx.x;
  const int lane = tid & 31;
  const int wave = tid >> 5;            // 0..2 -> M tile
  const int l15  = lane & 15;
  const int hi   = lane >> 4;
  const int kh   = hi << 1;

  const float* q = dw + ((size_t)b * 1152 +       head * 48) * 4096;
  const float* k = dw + ((size_t)b * 1152 + 384 + head * 48) * 4096;

  v8f acc[3] = {{}, {}, {}};
  const int mrow = wave * 16 + l15;

  for (int n = 0; n < 4096; n += 4) {
    const int kf = n + kh;
    v2f a;
    a.x = q[(size_t)mrow * 4096 + kf];
    a.y = q[(size_t)mrow * 4096 + kf + 1];
#pragma unroll
    for (int j = 0; j < 3; ++j) {
      v2f bb;  // B[kf][e] = k[e][kf]
      bb.x = k[(size_t)(j * 16 + l15) * 4096 + kf];
      bb.y = k[(size_t)(j * 16 + l15) * 4096 + kf + 1];
      acc[j] = WMMA_F32(a, bb, acc[j]);
    }
  }

  __shared__ float S[48][49];
  const float* invb = inv + b * 768;
  const float  t    = temp[head];
  const int    rb   = wave * 16 + hi * 8;
#pragma unroll
  for (int r = 0; r < 8; ++r) {
    const int   d  = rb + r;
    const float iq = invb[head * 48 + d];
#pragma unroll
    for (int j = 0; j < 3; ++j) {
      const int e = j * 16 + l15;
      S[d][e] = acc[j][r] * iq * invb[384 + head * 48 + e] * t;
    }
  }
  __syncthreads();

  if (tid < 48) {
    float mx = -3.402823466e+38f;
    for (int e = 0; e < 48; ++e) mx = fmaxf(mx, S[tid][e]);
    float sum = 0.0f;
    for (int e = 0; e < 48; ++e) {
      const float ev = expf(S[tid][e] - mx);
      S[tid][e] = ev;
      sum += ev;
    }
    const float rs = 1.0f / sum;
    for (int e = 0; e < 48; ++e) S[tid][e] *= rs;
  }
  __syncthreads();

  float* ao = attn + (size_t)bh * 48 * 48;
  for (int i = tid; i < 48 * 48; i += 96) ao[i] = S[i / 48][i % 48];
}

// ---------------------------------------------------------------------------
// out[b][head*48+d][n] = sum_e attn[bh][d][e] * v[b][head][e][n]
// Per block: one (b,head), 64-wide N strip. 3 waves (M tiles) x 4 N sub-tiles.
// grid = (64 n-strips, 64 bh), block = 96. Output compact [B][384][4096].
// ---------------------------------------------------------------------------
__global__ __launch_bounds__(96) void attn_v(
    const float* __restrict__ attn, const float* __restrict__ dw,
    float* __restrict__ out)
{
  const int bh   = blockIdx.y;
  const int b    = bh >> 3;
  const int head = bh & 7;
  const int n0   = blockIdx.x * 64;
  const int tid  = threadIdx.x;
  const int lane = tid & 31;
  const int wave = tid >> 5;            // M tile 0..2
  const int l15  = lane & 15;
  const int hi   = lane >> 4;
  const int kh   = hi << 1;

  const float* A = attn + (size_t)bh * 48 * 48;                       // [48][48]
  const float* V = dw + ((size_t)b * 1152 + 768 + head * 48) * 4096;  // [48][4096]

  v8f acc[4] = {{}, {}, {}, {}};
  const int mrow = wave * 16 + l15;

#pragma unroll
  for (int kk = 0; kk < 48; kk += 4) {
    const int kf = kk + kh;
    v2f a;
    a.x = A[mrow * 48 + kf];
    a.y = A[mrow * 48 + kf + 1];
#pragma unroll
    for (int j = 0; j < 4; ++j) {
      v2f bb;
      bb.x = V[(size_t)kf * 4096 + n0 + j * 16 + l15];
      bb.y = V[(size_t)(kf + 1) * 4096 + n0 + j * 16 + l15];
      acc[j] = WMMA_F32(a, bb, acc[j]);
    }
  }

  float* O = out + ((size_t)b * 384 + head * 48) * 4096;
  const int rb = wave * 16 + hi * 8;
#pragma unroll
  for (int r = 0; r < 8; ++r)
#pragma unroll
    for (int j = 0; j < 4; ++j)
      O[(size_t)(rb + r) * 4096 + n0 + j * 16 + l15] = acc[j][r];
}

// ---------------------------------------------------------------------------
// Host-side launcher. Inputs: x, qkv_w, dw_w, proj_w, temperature (all fp32).
// ---------------------------------------------------------------------------
extern "C" void kernel_launch(void* const* d_in, const int* in_sizes, int n_in,
                              void* d_out, int out_size, void* d_ws, size_t ws_size,
                              hipStream_t stream) {
  (void)in_sizes; (void)n_in; (void)out_size; (void)ws_size;

  const float* x      = (const float*)d_in[0];   // [8][384][64][64]
  const float* qkv_w  = (const float*)d_in[1];   // [1152][384]
  const float* dw_w   = (const float*)d_in[2];   // [1152][9]
  const float* proj_w = (const float*)d_in[3];   // [384][384]
  const float* temp   = (const float*)d_in[4];   // [8]
  float*       out    = (float*)d_out;           // [8][384][4096]

  char* ws = (char*)d_ws;
  const size_t big = (size_t)8 * 1152 * 4096 * sizeof(float);   // 151 MB
  float* buf0 = (float*)ws;                        // qkv pre-dw, then attn@v out
  float* buf1 = (float*)(ws + big);                // qkv post-dw
  float* inv  = (float*)(ws + 2 * big);            // [8][768]
  float* attn = (float*)(ws + 2 * big + (size_t)8 * 768 * sizeof(float)); // [64][48][48]

  // 1) qkv 1x1 conv as GEMM:  [1152x384] @ [384x4096] per batch
  gemm_wmma_f32<<<dim3(32, 18, 8), 256, 0, stream>>>(qkv_w, x, buf0, 1152, 384, 4096);

  // 2) depthwise 3x3
  dwconv3x3<<<(8 * 1152 * 4096) / 256, 256, 0, stream>>>(buf0, dw_w, buf1);

  // 3) L2 norms of q,k rows
  l2norm_rows<<<8 * 768, 256, 0, stream>>>(buf1, inv);

  // 4) attention matrix + softmax
  attn_softmax<<<64, 96, 0, stream>>>(buf1, inv, temp, attn);

  // 5) attn @ v  -> compact [8][384][4096] in buf0
  attn_v<<<dim3(64, 64), 96, 0, stream>>>(attn, buf1, buf0);

  // 6) projection 1x1 conv as GEMM:  [384x384] @ [384x4096] per batch
  gemm_wmma_f32<<<dim3(32, 6, 8), 256, 0, stream>>>(proj_w, buf0, out, 384, 384, 4096);
}